// BilinearDecoder_35055523070807
// MI455X (gfx1250) — compile-verified
//
#include <hip/hip_runtime.h>
#include <hip/hip_bf16.h>
#include <stdint.h>

#define DIMN 256
#define NREL 30
#define LSTRIDE 260   // padded row stride (floats): bank-conflict-free, 16B-aligned rows
#define NCHUNK (DIMN / 16)

typedef float v2f __attribute__((ext_vector_type(2)));
typedef float v8f __attribute__((ext_vector_type(8)));

// Async global -> LDS copy, 16B per lane, tracked by ASYNCcnt.
// vdst = per-lane LDS byte offset (low 32 bits of generic shared pointer),
// vaddr = 64-bit global address in a VGPR pair.
__device__ __forceinline__ void bd_async_b128(const void* gsrc, const void* ldst) {
    uint32_t lds_off = (uint32_t)(uintptr_t)ldst;
    uint64_t gaddr   = (uint64_t)(uintptr_t)gsrc;
    asm volatile("global_load_async_to_lds_b128 %0, %1, off"
                 :
                 : "v"(lds_off), "v"(gaddr)
                 : "memory");
}

// ---------------- binning kernels ----------------

__global__ void bd_init_kernel(int* counts, int* cursor) {
    int t = threadIdx.x;
    if (t < 32) { counts[t] = 0; cursor[t] = 0; }
}

__global__ void bd_count_kernel(const int* __restrict__ rel, int* __restrict__ counts, int n) {
    int b = blockIdx.x * blockDim.x + threadIdx.x;
    if (b < n) atomicAdd(&counts[rel[b]], 1);
}

__global__ void bd_scan_kernel(const int* __restrict__ counts, int* __restrict__ offsets) {
    if (threadIdx.x == 0) {
        int acc = 0;
        for (int r = 0; r < NREL; ++r) { offsets[r] = acc; acc += counts[r]; }
    }
}

__global__ void bd_scatter_kernel(const int* __restrict__ rel, const int* __restrict__ offsets,
                                  int* __restrict__ cursor, int* __restrict__ order, int n) {
    int b = blockIdx.x * blockDim.x + threadIdx.x;
    if (b < n) {
        int r = rel[b];
        int pos = offsets[r] + atomicAdd(&cursor[r], 1);
        order[pos] = b;
    }
}

// ---------------- main bilinear tile kernel ----------------
// grid: (ceil(B/16), NREL), block: 128 threads (4 waves)
// 16 same-relation rows; V = H(16x256) @ M_r(256x256) via v_wmma_f32_16x16x4_f32,
// M_r streamed through double-buffered LDS with global_load_async_to_lds_b128.

__launch_bounds__(128, 4)
__global__ void bd_bilinear_kernel(const float* __restrict__ head,
                                   const float* __restrict__ tail,
                                   const float* __restrict__ mats,
                                   const int* __restrict__ order,
                                   const int* __restrict__ counts,
                                   const int* __restrict__ offsets,
                                   float* __restrict__ out) {
    __shared__ float Hs[16 * LSTRIDE];
    __shared__ float Ts[16 * LSTRIDE];
    __shared__ float Ms[2][16 * LSTRIDE];
    __shared__ float Ss[16];
    __shared__ int   rows[16];

    const int r    = blockIdx.y;
    const int tile = blockIdx.x;
    const int cnt  = counts[r];
    if (tile * 16 >= cnt) return;              // uniform block-level early exit

    const int base = offsets[r] + tile * 16;
    const int tid  = threadIdx.x;
    const int lane = tid & 31;
    const int wave = tid >> 5;
    const int half = lane >> 4;                // 0: lanes 0-15, 1: lanes 16-31
    const int l16  = lane & 15;

    if (tid < 16) {
        int p = tile * 16 + tid;
        rows[tid] = (p < cnt) ? order[base + tid] : -1;
        Ss[tid] = 0.0f;
    }

    const float* Mr = mats + (size_t)r * DIMN * DIMN;

    // Issue async staging of M chunk 0 while we gather H/T.
    {
        const int c4 = tid & 63;
        const int r0 = tid >> 6;
        #pragma unroll
        for (int i = 0; i < 8; ++i) {
            const int row = r0 + 2 * i;
            bd_async_b128(Mr + (size_t)row * DIMN + c4 * 4,
                          &Ms[0][row * LSTRIDE + c4 * 4]);
        }
    }

    // Gather 16 H and T rows into LDS (zero-fill inactive rows so WMMA runs full-EXEC).
    __syncthreads();                            // rows[] visible
    for (int e = tid; e < 16 * 64; e += 128) {
        int m  = e >> 6;
        int c4 = e & 63;
        int b  = rows[m];
        float4 hv = make_float4(0.f, 0.f, 0.f, 0.f);
        float4 tv = make_float4(0.f, 0.f, 0.f, 0.f);
        if (b >= 0) {
            hv = ((const float4*)(head + (size_t)b * DIMN))[c4];
            tv = ((const float4*)(tail + (size_t)b * DIMN))[c4];
        }
        *(float4*)(Hs + m * LSTRIDE + c4 * 4) = hv;
        *(float4*)(Ts + m * LSTRIDE + c4 * 4) = tv;
    }

    const int nbase = wave * 64;               // this wave's N-column base

    v8f acc[4];
    acc[0] = (v8f){0,0,0,0,0,0,0,0};
    acc[1] = acc[0]; acc[2] = acc[0]; acc[3] = acc[0];

    auto compute_chunk = [&](const float* Mc, int kc) {
        #pragma unroll
        for (int ks = 0; ks < 4; ++ks) {
            const int kl = ks * 4 + 2 * half;  // this lane's K rows: kl, kl+1
            const int kg = kc * 16 + kl;       // global K for A
            v2f a;
            a.x = Hs[l16 * LSTRIDE + kg];
            a.y = Hs[l16 * LSTRIDE + kg + 1];
            #pragma unroll
            for (int j = 0; j < 4; ++j) {
                const int col = nbase + j * 16 + l16;
                v2f b;
                b.x = Mc[kl * LSTRIDE + col];
                b.y = Mc[(kl + 1) * LSTRIDE + col];
                acc[j] = __builtin_amdgcn_wmma_f32_16x16x4_f32(
                    /*neg_a=*/false, a, /*neg_b=*/false, b,
                    /*c_mod=*/(short)0, acc[j],
                    /*reuse_a=*/false, /*reuse_b=*/false);
            }
        }
    };

    // Double-buffered main loop: stage chunk kc+1 asynchronously while computing kc.
    for (int kc = 0; kc < NCHUNK - 1; ++kc) {
        {   // issue chunk kc+1 into the alternate buffer
            const int c4 = tid & 63;
            const int r0 = tid >> 6;
            float* dst = Ms[(kc + 1) & 1];
            const float* src = Mr + (size_t)(kc + 1) * 16 * DIMN;
            #pragma unroll
            for (int i = 0; i < 8; ++i) {
                const int row = r0 + 2 * i;
                bd_async_b128(src + (size_t)row * DIMN + c4 * 4,
                              dst + row * LSTRIDE + c4 * 4);
            }
        }
        __builtin_amdgcn_s_wait_asynccnt(8);   // chunk kc's 8 ops retired (in-order)
        __syncthreads();                        // publish chunk kc across waves
        compute_chunk(Ms[kc & 1], kc);
        __syncthreads();                        // all reads of buf[kc&1] done before re-issue
    }
    __builtin_amdgcn_s_wait_asynccnt(0);       // last chunk complete
    __syncthreads();
    compute_chunk(Ms[(NCHUNK - 1) & 1], NCHUNK - 1);

    // Epilogue: s[row] += sum_col V[row,col] * T[row,col] over this wave's 64 cols.
    float pr[8];
    #pragma unroll
    for (int g = 0; g < 8; ++g) pr[g] = 0.0f;
    #pragma unroll
    for (int j = 0; j < 4; ++j) {
        const int col = nbase + j * 16 + l16;
        #pragma unroll
        for (int g = 0; g < 8; ++g) {
            const int row = g + 8 * half;      // D-tile VGPR layout
            pr[g] += acc[j][g] * Ts[row * LSTRIDE + col];
        }
    }
    // Butterfly reduce across the 16 column-lanes of each half.
    #pragma unroll
    for (int off = 1; off < 16; off <<= 1) {
        #pragma unroll
        for (int g = 0; g < 8; ++g)
            pr[g] += __shfl_xor(pr[g], off, 32);
    }
    if (l16 == 0) {
        #pragma unroll
        for (int g = 0; g < 8; ++g)
            atomicAdd(&Ss[g + 8 * half], pr[g]);   // ds_add_f32, combine 4 waves
    }
    __syncthreads();
    if (tid < 16) {
        int b = rows[tid];
        if (b >= 0) out[b] = Ss[tid];
    }
}

// ---------------- launcher ----------------

extern "C" void kernel_launch(void* const* d_in, const int* in_sizes, int n_in,
                              void* d_out, int out_size, void* d_ws, size_t ws_size,
                              hipStream_t stream) {
    const float* head = (const float*)d_in[0];
    const int*   rel  = (const int*)d_in[1];
    const float* tail = (const float*)d_in[2];
    const float* mats = (const float*)d_in[3];
    float* out = (float*)d_out;

    const int n = in_sizes[1];                 // batch = 4096

    int* counts  = (int*)d_ws;                 // [32]
    int* cursor  = counts + 32;                // [32]
    int* offsets = cursor + 32;                // [32]
    int* order   = offsets + 32;               // [n]

    bd_init_kernel<<<1, 64, 0, stream>>>(counts, cursor);
    bd_count_kernel<<<(n + 255) / 256, 256, 0, stream>>>(rel, counts, n);
    bd_scan_kernel<<<1, 32, 0, stream>>>(counts, offsets);
    bd_scatter_kernel<<<(n + 255) / 256, 256, 0, stream>>>(rel, offsets, cursor, order, n);

    dim3 grid((n + 15) / 16, NREL);
    bd_bilinear_kernel<<<grid, 128, 0, stream>>>(head, tail, mats, order, counts, offsets, out);
}